// SpatialAttention_64484638982313
// MI455X (gfx1250) — compile-verified
//
#include <hip/hip_runtime.h>
#include <math.h>

// SpatialAttention on MI455X (gfx1250).
// Key algebraic rewrite: mask MULTIPLIES scores (off-band -> 0, softmaxed over the
// FULL row), so attn = rank-1 uniform term + 5-diagonal band term. This removes both
// O(B*N^2*C) dense matmuls. Remaining heavy op: k=Wk@X, v=Wv@X GEMMs done with
// V_WMMA_F32_16X16X4_F32 (fp32-exact). Whole pipeline (~84MB) fits in 192MB L2;
// HBM traffic ~50MB -> ~2.2us bound, so fp32 WMMA throughput is ample.

#define B_   4
#define CH   256
#define N_   4096
#define NEAR 2
#define BAND 5

typedef __attribute__((ext_vector_type(2))) float v2f;
typedef __attribute__((ext_vector_type(8))) float v8f;

// ---------------------------------------------------------------------------
// K1: k = Wk @ X, v = Wv @ X  (per batch: [256,256] x [256,4096])
// One wave computes one 16x16 output tile; K-loop of 64 x V_WMMA_F32_16X16X4_F32.
// A (16x4 f32) per ISA layout: lanes 0-15 carry K=0,1; lanes 16-31 carry K=2,3
// (contiguous -> single float2 load). B (4x16): row striped across lanes.
// ---------------------------------------------------------------------------
__global__ __launch_bounds__(256) void gemm_kv_kernel(
    const float* __restrict__ X,   // [B, CH, N]
    const float* __restrict__ Wk,  // [CH, CH]
    const float* __restrict__ Wv,  // [CH, CH]
    float* __restrict__ kbuf,      // [B, CH, N]
    float* __restrict__ vbuf)      // [B, CH, N]
{
    const int lane = threadIdx.x & 31;
    const int wave = threadIdx.x >> 5;

    int t = blockIdx.x * 8 + wave;       // tile id: B*2*16*256 = 32768 tiles
    const int nt    = t & 255;  t >>= 8; // 256 N-tiles
    const int mt    = t & 15;   t >>= 4; // 16  M-tiles
    const int which = t & 1;    t >>= 1; // 0 = Wk, 1 = Wv
    const int b     = t;

    const int n0 = nt * 16;
    const int r0 = mt * 16;
    const float* W   = which ? Wv : Wk;
    float* outp      = which ? vbuf : kbuf;

    const int half = lane >> 4;   // 0: lanes 0-15, 1: lanes 16-31
    const int l16  = lane & 15;

    // A element pair: W[r0+l16, c0 + 2*half + {0,1}]  (contiguous float2)
    const float* aptr = W + (size_t)(r0 + l16) * CH + 2 * half;
    // B element pair: X[b, c0 + 2*half + {0,1}, n0 + l16]
    const float* bptr = X + ((size_t)b * CH + 2 * half) * N_ + n0 + l16;

    v8f acc = {};
#pragma unroll 8
    for (int c0 = 0; c0 < CH; c0 += 4) {
        v2f a = *(const v2f*)aptr;
        v2f bb;
        bb.x = bptr[0];
        bb.y = bptr[N_];
        acc = __builtin_amdgcn_wmma_f32_16x16x4_f32(
            /*neg_a=*/false, a, /*neg_b=*/false, bb,
            /*c_mod=*/(short)0, acc, /*reuse_a=*/false, /*reuse_b=*/false);
        aptr += 4;
        bptr += 4 * (size_t)N_;
    }

    // D layout: VGPR j -> row (r0 + j + half*8), col n0 + l16
    float* op = outp + ((size_t)b * CH + r0 + half * 8) * N_ + n0 + l16;
#pragma unroll
    for (int j = 0; j < 8; ++j)
        op[(size_t)j * N_] = acc[j];
}

// ---------------------------------------------------------------------------
// K2: banded scores s[n, n+d] = scale * sum_r hidden[b,r,n]*k[b,r,n+d], d=-2..2,
// then the full-row softmax in closed form:
//   denom = sum_band exp(s - M) + (N - cnt) * exp(-M),  M = max(0, max_band s)
//   rinv  = exp(-M)/denom       (uniform off-band weight)
//   g[j]  = exp(s_j - M)/denom - rinv   (band correction coeff), 0 if m out of range
// One thread per (b, n); loads coalesced in n.
// ---------------------------------------------------------------------------
__global__ __launch_bounds__(256) void band_softmax_kernel(
    const float* __restrict__ hidden,  // [B, CH, N]
    const float* __restrict__ kbuf,    // [B, CH, N]
    float* __restrict__ gbuf,          // [B, BAND, N]
    float* __restrict__ rinv)          // [B, N]
{
    const int gid = blockIdx.x * 256 + threadIdx.x;  // B*N threads
    const int b = gid >> 12;
    const int n = gid & (N_ - 1);
    const float scale = 0.0625f;  // 1/sqrt(256)

    bool valid[BAND];
#pragma unroll
    for (int j = 0; j < BAND; ++j) {
        int m = n - NEAR + j;
        valid[j] = (m >= 0) && (m < N_);
    }

    const float* q  = hidden + ((size_t)b * CH) * N_ + n;
    const float* kc = kbuf   + ((size_t)b * CH) * N_ + n;

    float s[BAND] = {0.f, 0.f, 0.f, 0.f, 0.f};
    for (int r = 0; r < CH; ++r) {
        float qv = q[(size_t)r * N_];
        const float* kr = kc + (size_t)r * N_;
#pragma unroll
        for (int j = 0; j < BAND; ++j)
            if (valid[j]) s[j] += qv * kr[j - NEAR];
    }

    float mx = 0.0f;
    int cnt = 0;
#pragma unroll
    for (int j = 0; j < BAND; ++j) {
        if (valid[j]) { s[j] *= scale; mx = fmaxf(mx, s[j]); cnt++; }
    }

    const float eoff = __expf(-mx);
    float denom = (float)(N_ - cnt) * eoff;
    float e[BAND];
#pragma unroll
    for (int j = 0; j < BAND; ++j) {
        e[j] = valid[j] ? __expf(s[j] - mx) : 0.0f;
        denom += e[j];
    }
    const float inv  = 1.0f / denom;
    const float roff = eoff * inv;

    rinv[(size_t)b * N_ + n] = roff;
#pragma unroll
    for (int j = 0; j < BAND; ++j)
        gbuf[((size_t)b * BAND + j) * N_ + n] = valid[j] ? (e[j] * inv - roff) : 0.0f;
}

// ---------------------------------------------------------------------------
// K3: S[b,c] = sum_n v[b,c,n] * rinv[b,n]   (rank-1 uniform-softmax term)
// ---------------------------------------------------------------------------
__global__ __launch_bounds__(256) void rank1_kernel(
    const float* __restrict__ vbuf, const float* __restrict__ rinv,
    float* __restrict__ S)
{
    const int b = blockIdx.x >> 8;
    const int c = blockIdx.x & 255;
    const float* vp = vbuf + ((size_t)b * CH + c) * N_;
    const float* rp = rinv + (size_t)b * N_;

    float acc = 0.0f;
    for (int i = threadIdx.x; i < N_; i += 256)
        acc += vp[i] * rp[i];

    __shared__ float red[256];
    red[threadIdx.x] = acc;
    __syncthreads();
#pragma unroll
    for (int off = 128; off > 0; off >>= 1) {
        if (threadIdx.x < off) red[threadIdx.x] += red[threadIdx.x + off];
        __syncthreads();
    }
    if (threadIdx.x == 0) S[blockIdx.x] = red[0];
}

// ---------------------------------------------------------------------------
// K4: out[b,c,m] = input[b,c,m] + S[b,c] + sum_{d=-2..2} v[b,c,m+d]*g[b,2-d,m+d]
// ---------------------------------------------------------------------------
__global__ __launch_bounds__(256) void epilogue_kernel(
    const float* __restrict__ X, const float* __restrict__ vbuf,
    const float* __restrict__ gbuf, const float* __restrict__ S,
    float* __restrict__ out)
{
    const size_t gid = (size_t)blockIdx.x * 256 + threadIdx.x;  // B*CH*N threads
    const int m  = (int)(gid & (N_ - 1));
    const int bc = (int)(gid >> 12);
    const int b  = bc >> 8;

    float acc = S[bc];
    const float* vp = vbuf + (size_t)bc * N_;
    const float* gb = gbuf + (size_t)b * BAND * N_;
#pragma unroll
    for (int d = -NEAR; d <= NEAR; ++d) {
        int n = m + d;
        if (n >= 0 && n < N_)
            acc += vp[n] * gb[(size_t)(NEAR - d) * N_ + n];  // j = (m-n)+NEAR = NEAR-d
    }
    out[gid] = X[gid] + acc;
}

// ---------------------------------------------------------------------------
extern "C" void kernel_launch(void* const* d_in, const int* in_sizes, int n_in,
                              void* d_out, int out_size, void* d_ws, size_t ws_size,
                              hipStream_t stream) {
    const float* input  = (const float*)d_in[0];  // [4, 256, 4096]
    const float* hidden = (const float*)d_in[1];  // [4, 256, 4096]
    const float* Wk     = (const float*)d_in[2];  // [256, 256]
    const float* Wv     = (const float*)d_in[3];  // [256, 256]
    float* out          = (float*)d_out;          // [4, 256, 4096]

    // Workspace carve-up (~34 MB of floats), all fully rewritten each call.
    float* ws   = (float*)d_ws;
    float* kbuf = ws;                                   // B*CH*N
    float* vbuf = kbuf + (size_t)B_ * CH * N_;          // B*CH*N
    float* gbuf = vbuf + (size_t)B_ * CH * N_;          // B*BAND*N
    float* rinv = gbuf + (size_t)B_ * BAND * N_;        // B*N
    float* Sbuf = rinv + (size_t)B_ * N_;               // B*CH

    // K1: 32768 wave-tiles / 8 waves per block
    gemm_kv_kernel<<<4096, 256, 0, stream>>>(input, Wk, Wv, kbuf, vbuf);
    // K2: one thread per (b, n)
    band_softmax_kernel<<<(B_ * N_) / 256, 256, 0, stream>>>(hidden, kbuf, gbuf, rinv);
    // K3: one block per (b, c)
    rank1_kernel<<<B_ * CH, 256, 0, stream>>>(vbuf, rinv, Sbuf);
    // K4: one thread per output element
    epilogue_kernel<<<(B_ * CH * N_) / 256, 256, 0, stream>>>(input, vbuf, gbuf, Sbuf, out);
}